// SimpleRNN_71906342470286
// MI455X (gfx1250) — compile-verified
//
#include <hip/hip_runtime.h>

#define T_STEPS 1000
#define IN_F 22
#define HID 10
#define NCLS 4
#define BATCH 4096

typedef float v2f __attribute__((ext_vector_type(2)));
typedef float v8f __attribute__((ext_vector_type(8)));

// Swap lane halves (lane <-> lane^16) via ds_swizzle SWAPX16 (xor=0x10,and=0x1f).
__device__ __forceinline__ float swap16(float v) {
  return __int_as_float(__builtin_amdgcn_ds_swizzle(__float_as_int(v), 0x401F));
}

// D = A(16x4 f32) * B(4x16 f32) + C, fp32 accumulate.
__device__ __forceinline__ v8f wmma4(v2f a, v2f b, v8f c) {
  return __builtin_amdgcn_wmma_f32_16x16x4_f32(false, a, false, b, (short)0, c,
                                               false, false);
}

// Load one K-chunk of a weight matrix into WMMA A-operand layout.
// A layout (f32 16x4): lanes 0-15 = M, lo-lanes hold K=4c,4c+1; hi-lanes K=4c+2,4c+3.
__device__ __forceinline__ v2f load_a(const float* __restrict__ W, int rows,
                                      int cols, int chunk, int lane) {
  int m = lane & 15;
  int k = chunk * 4 + ((lane & 16) ? 2 : 0);
  v2f a;
  a.x = (m < rows && k < cols) ? W[m * cols + k] : 0.f;
  a.y = (m < rows && (k + 1) < cols) ? W[m * cols + k + 1] : 0.f;
  return a;
}

// Build the three K-chunk B operands (rows = hidden unit 0..11, cols = batch)
// from an H^T tile held in C/D accumulator layout.
// B layout (f32 4x16): V0 = row 4c (lo lanes) / 4c+2 (hi lanes); V1 = 4c+1 / 4c+3.
__device__ __forceinline__ void make_b_from_h(const v8f& h, bool hi, v2f bb[3]) {
  float s0 = swap16(h[0]);  // rows 8 (as seen by lo lanes)
  float s1 = swap16(h[1]);  // row 9
  float s2 = swap16(h[2]);  // row 2 (as seen by hi lanes)
  float s3 = swap16(h[3]);
  float s6 = swap16(h[6]);
  float s7 = swap16(h[7]);
  bb[0].x = hi ? s2 : h[0];   // rows 0 / 2
  bb[0].y = hi ? s3 : h[1];   // rows 1 / 3
  bb[1].x = hi ? s6 : h[4];   // rows 4 / 6
  bb[1].y = hi ? s7 : h[5];   // rows 5 / 7
  bb[2].x = hi ? 0.f : s0;    // rows 8 / 10(pad)
  bb[2].y = hi ? 0.f : s1;    // rows 9 / 11(pad)
}

__device__ __forceinline__ v8f relu8(v8f v) {
  v8f r;
#pragma unroll
  for (int i = 0; i < 8; ++i) r[i] = fmaxf(v[i], 0.f);
  return r;
}

__global__ void __launch_bounds__(32)
rnn2_fused_wmma(const float* __restrict__ x,
                const float* __restrict__ W_ih0, const float* __restrict__ b_ih0,
                const float* __restrict__ W_hh0, const float* __restrict__ b_hh0,
                const float* __restrict__ W_ih1, const float* __restrict__ b_ih1,
                const float* __restrict__ W_hh1, const float* __restrict__ b_hh1,
                const float* __restrict__ W_lin, const float* __restrict__ b_lin,
                float* __restrict__ out) {
  const int lane = threadIdx.x & 31;
  const bool hi = (lane & 16) != 0;
  const int bidx = (blockIdx.x << 4) + (lane & 15);  // batch column for this lane
  const float* __restrict__ xrow = x + (size_t)bidx * (T_STEPS * IN_F);

  // ---- loop-invariant weight tiles in A layout ----
  v2f aih0[6], ahh0[3], aih1[3], ahh1[3], alin[3];
#pragma unroll
  for (int c = 0; c < 6; ++c) aih0[c] = load_a(W_ih0, HID, IN_F, c, lane);
#pragma unroll
  for (int c = 0; c < 3; ++c) {
    ahh0[c] = load_a(W_hh0, HID, HID, c, lane);
    aih1[c] = load_a(W_ih1, HID, HID, c, lane);
    ahh1[c] = load_a(W_hh1, HID, HID, c, lane);
    alin[c] = load_a(W_lin, NCLS, HID, c, lane);
  }

  // ---- bias tiles in C/D layout: VGPR v holds row v (lo lanes) / v+8 (hi lanes) ----
  v8f bias0, bias1, biasL;
#pragma unroll
  for (int v = 0; v < 8; ++v) {
    int m = v + (hi ? 8 : 0);
    bias0[v] = (m < HID) ? (b_ih0[m] + b_hh0[m]) : 0.f;
    bias1[v] = (m < HID) ? (b_ih1[m] + b_hh1[m]) : 0.f;
    biasL[v] = (m < NCLS) ? b_lin[m] : 0.f;
  }

  v8f h0 = {};  // H0^T tile, C/D layout (zero initial state)
  v8f h1 = {};  // H1^T tile

  // ---- x-tile B operands for t = 0 (rows = input feature, cols = batch) ----
  const int k0 = hi ? 2 : 0;
  v2f xb[6];
#pragma unroll
  for (int c = 0; c < 6; ++c) {
    int k = 4 * c + k0;
    xb[c].x = (k < IN_F) ? xrow[k] : 0.f;
    xb[c].y = (k + 1 < IN_F) ? xrow[k + 1] : 0.f;
  }

  for (int t = 0; t < T_STEPS; ++t) {
    // prefetch next timestep's x tile (off critical path)
    int tn = (t + 1 < T_STEPS) ? (t + 1) : t;
    const float* __restrict__ xr = xrow + tn * IN_F;
    v2f xbn[6];
#pragma unroll
    for (int c = 0; c < 6; ++c) {
      int k = 4 * c + k0;
      xbn[c].x = (k < IN_F) ? xr[k] : 0.f;
      xbn[c].y = (k + 1 < IN_F) ? xr[k + 1] : 0.f;
    }

    // layer 0: acc = b0 + W_ih0 * x_t^T   (independent of recurrence)
    v8f acc = bias0;
#pragma unroll
    for (int c = 0; c < 6; ++c) acc = wmma4(aih0[c], xb[c], acc);
    // + W_hh0 * h0^T   (recurrence critical path)
    v2f bh[3];
    make_b_from_h(h0, hi, bh);
#pragma unroll
    for (int c = 0; c < 3; ++c) acc = wmma4(ahh0[c], bh[c], acc);
    h0 = relu8(acc);

    // layer 1: acc1 = b1 + W_ih1 * h0_new^T + W_hh1 * h1^T
    v8f acc1 = bias1;
    v2f bh1[3];
    make_b_from_h(h1, hi, bh1);
#pragma unroll
    for (int c = 0; c < 3; ++c) acc1 = wmma4(ahh1[c], bh1[c], acc1);
    v2f bh0n[3];
    make_b_from_h(h0, hi, bh0n);
#pragma unroll
    for (int c = 0; c < 3; ++c) acc1 = wmma4(aih1[c], bh0n[c], acc1);
    h1 = relu8(acc1);

#pragma unroll
    for (int c = 0; c < 6; ++c) xb[c] = xbn[c];
  }

  // final head: logits^T = W_lin * h1^T + b_lin   (h1 already post-relu)
  v8f lg = biasL;
  v2f bhl[3];
  make_b_from_h(h1, hi, bhl);
#pragma unroll
  for (int c = 0; c < 3; ++c) lg = wmma4(alin[c], bhl[c], lg);

  if (!hi) {
    float* __restrict__ o = out + (size_t)bidx * NCLS;
#pragma unroll
    for (int v = 0; v < NCLS; ++v) o[v] = lg[v];
  }
}

extern "C" void kernel_launch(void* const* d_in, const int* in_sizes, int n_in,
                              void* d_out, int out_size, void* d_ws, size_t ws_size,
                              hipStream_t stream) {
  (void)in_sizes; (void)n_in; (void)out_size; (void)d_ws; (void)ws_size;
  const float* x     = (const float*)d_in[0];
  const float* W_ih0 = (const float*)d_in[1];
  const float* b_ih0 = (const float*)d_in[2];
  const float* W_hh0 = (const float*)d_in[3];
  const float* b_hh0 = (const float*)d_in[4];
  const float* W_ih1 = (const float*)d_in[5];
  const float* b_ih1 = (const float*)d_in[6];
  const float* W_hh1 = (const float*)d_in[7];
  const float* b_hh1 = (const float*)d_in[8];
  const float* W_lin = (const float*)d_in[9];
  const float* b_lin = (const float*)d_in[10];
  float* out = (float*)d_out;

  dim3 grid(BATCH / 16);  // one wave per 16-batch tile -> 256 waves
  dim3 block(32);
  hipLaunchKernelGGL(rnn2_fused_wmma, grid, block, 0, stream,
                     x, W_ih0, b_ih0, W_hh0, b_hh0,
                     W_ih1, b_ih1, W_hh1, b_hh1, W_lin, b_lin, out);
}